// TagAgnosticGenerator_69801808495259
// MI455X (gfx1250) — compile-verified
//
#include <hip/hip_runtime.h>
#include <hip/hip_bf16.h>

typedef __attribute__((ext_vector_type(2))) float v2f;
typedef __attribute__((ext_vector_type(8))) float v8f;

#define EPS 1e-20f
#define NT 7            // N tiles of 16 covering D=100 (<=112)
#define K2_WAVES 8      // waves in kernel 2

struct RowRec {
    float m2;       // running max of t = x + gumbel for this row
    float inv_s2;   // 1 / sum(exp(t - m2))
    int   sel;      // 1 = use sampled_emb (masked & safe), 0 = gather W[gidx]
    int   gidx;     // inp_word (unmasked) or UNK_ID=0 (masked & unsafe)
};

__global__ void init_kernel(float* ent_sum, int* n_masked) {
    *ent_sum = 0.0f;
    *n_masked = 0;
}

// One block per (b,s) row: streaming online reductions over V elements.
__global__ __launch_bounds__(256) void row_stats_kernel(
    const float* __restrict__ logits, const float* __restrict__ gumbel,
    const long long* __restrict__ inp_word, const int* __restrict__ obf_mask,
    float* __restrict__ out_word, RowRec* __restrict__ recs,
    float* __restrict__ ent_sum, int* __restrict__ n_masked, int V)
{
    const int row = blockIdx.x;
    const int tid = threadIdx.x;
    const float* xr = logits + (size_t)row * V;
    const float* ur = gumbel + (size_t)row * V;

    float m = -INFINITY, S = 0.f, XS = 0.f;   // max, sumexp, sum(x*exp(x-m))
    float m2 = -INFINITY, S2 = 0.f;           // same for t = x + g
    float bt = -INFINITY; int bi = 0x7fffffff;

    for (int v = tid; v < V; v += 256) {
        float x = xr[v];
        float u = ur[v];
        float g = -__logf(-__logf(u + EPS) + EPS);
        float t = x + g;
        if (x > m) { float r = __expf(m - x); S *= r; XS *= r; m = x; }
        float e = __expf(x - m);
        S += e; XS += x * e;
        if (t > m2) { S2 *= __expf(m2 - t); m2 = t; }
        S2 += __expf(t - m2);
        if (t > bt) { bt = t; bi = v; }   // first (lowest v) kept on ties
    }

    __shared__ float sm[256], sS[256], sXS[256], sm2[256], sS2[256], sbt[256];
    __shared__ int sbi[256];
    sm[tid] = m; sS[tid] = S; sXS[tid] = XS;
    sm2[tid] = m2; sS2[tid] = S2; sbt[tid] = bt; sbi[tid] = bi;
    __syncthreads();

    for (int off = 128; off > 0; off >>= 1) {
        if (tid < off) {
            float ma = sm[tid], mb = sm[tid + off];
            float M = fmaxf(ma, mb);
            float ea = __expf(ma - M), eb = __expf(mb - M);
            sm[tid]  = M;
            sS[tid]  = sS[tid]  * ea + sS[tid + off]  * eb;
            sXS[tid] = sXS[tid] * ea + sXS[tid + off] * eb;

            float m2a = sm2[tid], m2b = sm2[tid + off];
            float M2 = fmaxf(m2a, m2b);
            sm2[tid] = M2;
            sS2[tid] = sS2[tid] * __expf(m2a - M2) + sS2[tid + off] * __expf(m2b - M2);

            float bta = sbt[tid], btb = sbt[tid + off];
            int bia = sbi[tid], bib = sbi[tid + off];
            if (btb > bta || (btb == bta && bib < bia)) { sbt[tid] = btb; sbi[tid] = bib; }
        }
        __syncthreads();
    }

    if (tid == 0) {
        float M = sm[0], Sf = sS[0], XSf = sXS[0];
        float lse = M + __logf(Sf);
        float ent = XSf / Sf - lse;               // sum(lp * exp(lp)) for this row
        int mask = obf_mask[row];
        long long iw = inp_word[row];
        int sidx = sbi[0];
        bool safe = ((long long)sidx != iw);
        long long ow = mask ? (safe ? (long long)sidx : 0LL) : iw;
        out_word[row] = (float)ow;
        if (mask) { atomicAdd(ent_sum, ent); atomicAdd(n_masked, 1); }
        RowRec rr;
        rr.m2 = sm2[0];
        rr.inv_s2 = 1.0f / sS2[0];
        rr.sel = (mask && safe) ? 1 : 0;
        rr.gidx = mask ? 0 : (int)iw;             // UNK_ID = 0
        recs[row] = rr;
    }
}

// One block per 16-row group. 8 waves split the K=V dimension; each wave
// accumulates 7 C tiles (16x16 f32) with v_wmma_f32_16x16x4_f32, partials
// reduced through LDS, then per-row select/scale on store.
//
// A fragment (16x4 f32): lane L holds row L%16, K = k0 + 2*(L>=16) + {0,1}
// B fragment (4x16 f32): lane L holds col n*16 + L%16, same K pair as A.
__global__ __launch_bounds__(256) void sampled_emb_wmma_kernel(
    const float* __restrict__ logits, const float* __restrict__ gumbel,
    const float* __restrict__ W, const RowRec* __restrict__ recs,
    float* __restrict__ out_emb, int V, int D)
{
    const int group = blockIdx.x;
    const int tid  = threadIdx.x;
    const int lane = tid & 31;
    const int wave = tid >> 5;
    const int half = (lane >> 4) & 1;
    const int r    = lane & 15;
    const int rowA = group * 16 + r;

    const float* xr = logits + (size_t)rowA * V;
    const float* ur = gumbel + (size_t)rowA * V;
    const float m2r = recs[rowA].m2;

    v8f c[NT] = {};

    const int kPerWave = V / K2_WAVES;           // 4000, multiple of 4
    const int k0beg = wave * kPerWave;
    const int k0end = k0beg + kPerWave;
    for (int k0 = k0beg; k0 < k0end; k0 += 4) {
        const int kA = k0 + half * 2;
        float x0 = xr[kA],     x1 = xr[kA + 1];
        float u0 = ur[kA],     u1 = ur[kA + 1];
        float g0 = -__logf(-__logf(u0 + EPS) + EPS);
        float g1 = -__logf(-__logf(u1 + EPS) + EPS);
        v2f a;
        a.x = __expf(x0 + g0 - m2r);             // unnormalized relaxed prob
        a.y = __expf(x1 + g1 - m2r);
        const float* Wk0 = W + (size_t)kA * D;
        const float* Wk1 = Wk0 + D;
#pragma unroll
        for (int n = 0; n < NT; ++n) {
            int col  = n * 16 + r;
            int colc = col < D ? col : (D - 1);  // clamp; masked at store
            v2f b;
            b.x = Wk0[colc];
            b.y = Wk1[colc];
            c[n] = __builtin_amdgcn_wmma_f32_16x16x4_f32(
                false, a, false, b, (short)0, c[n], false, false);
        }
    }

    // Cross-wave reduction of partial C tiles.
    __shared__ float red[K2_WAVES][32][NT * 8];  // 57344 B
#pragma unroll
    for (int n = 0; n < NT; ++n)
#pragma unroll
        for (int i = 0; i < 8; ++i)
            red[wave][lane][n * 8 + i] = c[n][i];
    __syncthreads();

    // C layout: VGPR i, lane L -> (row = i + 8*(L>=16), col = n*16 + L%16)
    for (int j = tid; j < 32 * NT * 8; j += 256) {
        int l = j / (NT * 8), reg = j % (NT * 8);
        float s = 0.f;
#pragma unroll
        for (int w = 0; w < K2_WAVES; ++w) s += red[w][l][reg];
        int n = reg / 8, i = reg % 8;
        int col = n * 16 + (l & 15);
        int rr  = i + ((l >> 4) ? 8 : 0);
        if (col < D) {
            int grow = group * 16 + rr;
            RowRec rec = recs[grow];
            float val = rec.sel ? (s * rec.inv_s2)
                                : W[(size_t)rec.gidx * D + col];
            out_emb[(size_t)grow * D + col] = val;
        }
    }
}

__global__ void finalize_kernel(const float* ent_sum, const int* n_masked,
                                float* out_ent, int V) {
    *out_ent = *ent_sum / ((float)(*n_masked) * (float)V);
}

extern "C" void kernel_launch(void* const* d_in, const int* in_sizes, int n_in,
                              void* d_out, int out_size, void* d_ws, size_t ws_size,
                              hipStream_t stream) {
    const float*     logits   = (const float*)d_in[0];
    const float*     gumbel   = (const float*)d_in[1];
    const long long* inp_word = (const long long*)d_in[2];
    const int*       obf_mask = (const int*)d_in[3];
    const float*     W        = (const float*)d_in[4];

    const int rows = in_sizes[2];            // 8*256 = 2048
    const int V    = in_sizes[0] / rows;     // 32000
    const int D    = in_sizes[4] / V;        // 100

    float* out      = (float*)d_out;
    float* out_word = out;                         // [rows]
    float* out_emb  = out + rows;                  // [rows*D]
    float* out_ent  = out + rows + (size_t)rows * D;

    float*  ent_sum  = (float*)d_ws;
    int*    n_masked = (int*)((char*)d_ws + 4);
    RowRec* recs     = (RowRec*)((char*)d_ws + 16);

    init_kernel<<<1, 1, 0, stream>>>(ent_sum, n_masked);
    row_stats_kernel<<<rows, 256, 0, stream>>>(logits, gumbel, inp_word, obf_mask,
                                               out_word, recs, ent_sum, n_masked, V);
    sampled_emb_wmma_kernel<<<rows / 16, 256, 0, stream>>>(logits, gumbel, W, recs,
                                                           out_emb, V, D);
    finalize_kernel<<<1, 1, 0, stream>>>(ent_sum, n_masked, out_ent, V);
}